// SSIMLoss_19499151524442
// MI455X (gfx1250) — compile-verified
//
#include <hip/hip_runtime.h>

// ---------------------------------------------------------------------------
// SSIM loss, fused single pass per 16x16 tile.
// Blurs run on the matrix pipe via V_WMMA_F32_16X16X4_F32 (banded-matrix
// formulation of the separable 11-tap Gaussian). HBM floor ~5.8us @23.3TB/s.
// ---------------------------------------------------------------------------

typedef __attribute__((ext_vector_type(2))) float v2f;
typedef __attribute__((ext_vector_type(8))) float v8f;

#define H_IMG 512
#define W_IMG 512
#define NPLANES 64                    // 32 batch * 2 channels
#define TILE 16
#define TILES_X (W_IMG / TILE)        // 32
#define TILES_Y (H_IMG / TILE)        // 32
#define NUM_TILES (NPLANES * TILES_X * TILES_Y)   // 65536
#define WPB 4                         // waves per block (wave32)
#define NBLOCKS (NUM_TILES / WPB)     // 16384

// per-wave LDS layout (floats)
#define PSTR 27                       // 26x26 patch, stride 27 (bank stagger)
#define PATCH_F (26 * PSTR)           // 702
#define HSTR 17                       // H-buffer 26x16, stride 17
#define HMAP_F (26 * HSTR)            // 442
#define HBUF_F (5 * HMAP_F)           // 2210
#define WAVE_F (2 * PATCH_F + HBUF_F) // 3614 floats = 14456 B per wave

#define NELEM_INV (1.0f / 16777216.0f)

// Normalized Gaussian(sigma=1.5, k=11) taps; w[i]=0 outside [0,10].
__device__ __forceinline__ float wsel(int i) {
  float r = 0.0f;
  r = (i == 0 || i == 10) ? 0.00102838f : r;
  r = (i == 1 || i == 9)  ? 0.00759876f : r;
  r = (i == 2 || i == 8)  ? 0.03600070f : r;
  r = (i == 3 || i == 7)  ? 0.10936060f : r;
  r = (i == 4 || i == 6)  ? 0.21300570f : r;
  r = (i == 5)            ? 0.26601160f : r;
  return r;
}

__device__ __forceinline__ v8f vzero8() {
  v8f v = {0.f, 0.f, 0.f, 0.f, 0.f, 0.f, 0.f, 0.f};
  return v;
}

__device__ __forceinline__ v8f wmma4(v2f a, v2f b, v8f c) {
  // 8 args: (neg_a, A, neg_b, B, c_mod, C, reuse_a, reuse_b)
  return __builtin_amdgcn_wmma_f32_16x16x4_f32(false, a, false, b, (short)0, c,
                                               false, false);
}

__global__ __launch_bounds__(WPB * 32) void ssim_tile_kernel(
    const float* __restrict__ x, const float* __restrict__ y,
    float* __restrict__ partial, float* __restrict__ accum, int use_partial) {
  extern __shared__ float smem[];
  const int lane = threadIdx.x & 31;
  const int wave = threadIdx.x >> 5;
  const int lm = lane & 15;           // N (or M) index within fragment
  const int hi = (lane >> 4) & 1;     // lane-half selects K pair {0,1}/{2,3}

  float* wbase = smem + wave * WAVE_F;
  float* xp = wbase;                  // 26x26 x patch (stride 27)
  float* yp = wbase + PATCH_F;        // 26x26 y patch
  float* hb = wbase + 2 * PATCH_F;    // 5 maps of 26x16 H-blurred (stride 17)
  float* wsum = smem + WPB * WAVE_F;  // WPB floats for block reduce

  // ---- tile coordinates -------------------------------------------------
  const int gw = blockIdx.x * WPB + wave;           // global wave == tile id
  const int plane = gw >> 10;                       // / (32*32)
  const int ty = (gw >> 5) & 31;
  const int tx = gw & 31;
  const int row0 = ty * TILE, col0 = tx * TILE;
  const float* xs = x + (size_t)plane * (H_IMG * W_IMG);
  const float* ys = y + (size_t)plane * (H_IMG * W_IMG);

  // ---- stage 26x26 halo patches into LDS (coalesced, intra-wave) --------
  for (int idx = lane; idx < 26 * 26; idx += 32) {
    const int r = idx / 26;
    const int c = idx - r * 26;
    const int gr = row0 + r - 5;
    const int gc = col0 + c - 5;
    float xv = 0.f, yv = 0.f;
    if (gr >= 0 && gr < H_IMG && gc >= 0 && gc < W_IMG) {
      const size_t o = (size_t)gr * W_IMG + gc;
      xv = xs[o];
      yv = ys[o];
    }
    xp[r * PSTR + c] = xv;
    yp[r * PSTR + c] = yv;
  }
  // LDS ops are in-order within a wave; no barrier needed (compiler waits DScnt).

  // ---- per-lane banded Gaussian fragments (shared by both passes) -------
  v2f wf[7];
#pragma unroll
  for (int q = 0; q < 7; ++q) {
    const int base = 4 * q + 2 * hi - lm;   // w[K - index], zero outside band
    wf[q].x = wsel(base);
    wf[q].y = wsel(base + 1);
  }

  // ---- horizontal pass: H[m] (26 rows x 16 cols), two 16-row halves -----
#pragma unroll
  for (int h = 0; h < 2; ++h) {
    v8f a0 = vzero8(), a1 = vzero8(), a2 = vzero8(), a3 = vzero8(),
        a4 = vzero8();
    const int r = h * 16 + lm;                      // this lane's A-row
    const float rmask = (r < 26) ? 1.f : 0.f;
    const int rc = (r < 26) ? r : 25;
#pragma unroll
    for (int q = 0; q < 7; ++q) {
      const int c0 = 4 * q + 2 * hi;
      const int c1 = c0 + 1;
      const float m0 = (c0 < 26) ? rmask : 0.f;
      const float m1 = (c1 < 26) ? rmask : 0.f;
      const int cc0 = (c0 < 26) ? c0 : 25;
      const int cc1 = (c1 < 26) ? c1 : 25;
      const float xv0 = xp[rc * PSTR + cc0] * m0;
      const float yv0 = yp[rc * PSTR + cc0] * m0;
      const float xv1 = xp[rc * PSTR + cc1] * m1;
      const float yv1 = yp[rc * PSTR + cc1] * m1;
      const v2f B = wf[q];
      v2f A;
      A.x = xv0;        A.y = xv1;        a0 = wmma4(A, B, a0);  // x
      A.x = yv0;        A.y = yv1;        a1 = wmma4(A, B, a1);  // y
      A.x = xv0 * xv0;  A.y = xv1 * xv1;  a2 = wmma4(A, B, a2);  // xx
      A.x = yv0 * yv0;  A.y = yv1 * yv1;  a3 = wmma4(A, B, a3);  // yy
      A.x = xv0 * yv0;  A.y = xv1 * yv1;  a4 = wmma4(A, B, a4);  // xy
    }
    // D layout: c[i] -> row (h*16 + i + 8*hi), col lm. Keep rows < 26.
#pragma unroll
    for (int i = 0; i < 8; ++i) {
      const int rr = h * 16 + i + hi * 8;
      if (rr < 26) {
        const int o = rr * HSTR + lm;
        hb[0 * HMAP_F + o] = a0[i];
        hb[1 * HMAP_F + o] = a1[i];
        hb[2 * HMAP_F + o] = a2[i];
        hb[3 * HMAP_F + o] = a3[i];
        hb[4 * HMAP_F + o] = a4[i];
      }
    }
  }

  // ---- vertical pass: V[m] = banded(W) x H[m] ---------------------------
  v8f u0 = vzero8(), u1 = vzero8(), u2 = vzero8(), u3 = vzero8(),
      u4 = vzero8();
#pragma unroll
  for (int q = 0; q < 7; ++q) {
    const int k0 = 4 * q + 2 * hi;
    const int k1 = k0 + 1;
    const float m0 = (k0 < 26) ? 1.f : 0.f;
    const float m1 = (k1 < 26) ? 1.f : 0.f;
    const int o0 = ((k0 < 26) ? k0 : 25) * HSTR + lm;
    const int o1 = ((k1 < 26) ? k1 : 25) * HSTR + lm;
    const v2f A = wf[q];
    v2f B;
    B.x = hb[0 * HMAP_F + o0] * m0; B.y = hb[0 * HMAP_F + o1] * m1;
    u0 = wmma4(A, B, u0);
    B.x = hb[1 * HMAP_F + o0] * m0; B.y = hb[1 * HMAP_F + o1] * m1;
    u1 = wmma4(A, B, u1);
    B.x = hb[2 * HMAP_F + o0] * m0; B.y = hb[2 * HMAP_F + o1] * m1;
    u2 = wmma4(A, B, u2);
    B.x = hb[3 * HMAP_F + o0] * m0; B.y = hb[3 * HMAP_F + o1] * m1;
    u3 = wmma4(A, B, u3);
    B.x = hb[4 * HMAP_F + o0] * m0; B.y = hb[4 * HMAP_F + o1] * m1;
    u4 = wmma4(A, B, u4);
  }

  // ---- SSIM epilogue on accumulator fragments ---------------------------
  const float C1 = 1e-4f;   // 0.01^2
  const float C2 = 9e-4f;   // 0.03^2
  float local = 0.f;
#pragma unroll
  for (int i = 0; i < 8; ++i) {
    const float ux = u0[i], uy = u1[i], uxx = u2[i], uyy = u3[i], uxy = u4[i];
    const float vx = uxx - ux * ux;
    const float vy = uyy - uy * uy;
    const float vxy = uxy - ux * uy;
    const float num = (2.f * ux * uy + C1) * (2.f * vxy + C2);
    const float den = (ux * ux + uy * uy + C1) * (vx + vy + C2);
    local += num / (den + 1e-12f);
  }

  // ---- wave32 reduce, then block reduce ---------------------------------
#pragma unroll
  for (int off = 16; off > 0; off >>= 1) local += __shfl_xor(local, off, 32);
  if (lane == 0) wsum[wave] = local;
  __syncthreads();
  if (threadIdx.x == 0) {
    float s = 0.f;
#pragma unroll
    for (int i = 0; i < WPB; ++i) s += wsum[i];
    if (use_partial)
      partial[blockIdx.x] = s;
    else
      atomicAdd(accum, s);
  }
}

__global__ void ssim_reduce_kernel(const float* __restrict__ partial,
                                   float* __restrict__ out) {
  __shared__ float sm[256];
  float s = 0.f;
  for (int i = threadIdx.x; i < NBLOCKS; i += 256) s += partial[i];
  sm[threadIdx.x] = s;
  __syncthreads();
  for (int off = 128; off > 0; off >>= 1) {
    if (threadIdx.x < off) sm[threadIdx.x] += sm[threadIdx.x + off];
    __syncthreads();
  }
  if (threadIdx.x == 0) out[0] = 1.0f - sm[0] * NELEM_INV;
}

__global__ void ssim_zero_kernel(float* p) {
  if (threadIdx.x == 0 && blockIdx.x == 0) p[0] = 0.f;
}

__global__ void ssim_finalize_kernel(float* p) {
  if (threadIdx.x == 0 && blockIdx.x == 0) p[0] = 1.0f - p[0] * NELEM_INV;
}

extern "C" void kernel_launch(void* const* d_in, const int* in_sizes, int n_in,
                              void* d_out, int out_size, void* d_ws,
                              size_t ws_size, hipStream_t stream) {
  const float* x = (const float*)d_in[0];
  const float* y = (const float*)d_in[1];
  float* out = (float*)d_out;
  const size_t smem = (size_t)(WPB * WAVE_F + WPB) * sizeof(float);

  if (ws_size >= (size_t)NBLOCKS * sizeof(float)) {
    // Deterministic two-stage reduction through workspace partials.
    float* partial = (float*)d_ws;
    ssim_tile_kernel<<<NBLOCKS, WPB * 32, smem, stream>>>(x, y, partial,
                                                          nullptr, 1);
    ssim_reduce_kernel<<<1, 256, 0, stream>>>(partial, out);
  } else {
    // Fallback: atomic accumulation directly into d_out.
    ssim_zero_kernel<<<1, 32, 0, stream>>>(out);
    ssim_tile_kernel<<<NBLOCKS, WPB * 32, smem, stream>>>(x, y, nullptr, out,
                                                          0);
    ssim_finalize_kernel<<<1, 32, 0, stream>>>(out);
  }
}